// StealNMSLossOld_21603685499337
// MI455X (gfx1250) — compile-verified
//
#include <hip/hip_runtime.h>
#include <stdint.h>

// ---------------- problem constants (from reference shapes) ----------------
#define H_IMG 512
#define W_IMG 1024
#define C_TOTAL 16
#define TH 16
#define TW 64
#define HS (TH + 4)          // 20 rows incl. halo 2
#define WS (TW + 4)          // 68 cols incl. halo 2
#define HALO_N (HS * WS)     // 1360
#define NTHREADS 256
#define CG 4                 // channels resident in LDS at a time

typedef __attribute__((ext_vector_type(2))) float v2f;
typedef __attribute__((ext_vector_type(8))) float v8f;
typedef unsigned long long u64;

__device__ __forceinline__ int clampi(int x, int lo, int hi) {
  return x < lo ? lo : (x > hi ? hi : x);
}

// Low 32 bits of a generic pointer to __shared__ == wave-relative LDS byte offset
// (generic LDS address = {SHARED_BASE[63:32], offset[31:0]} per CDNA5 aperture rules).
__device__ __forceinline__ unsigned lds_off(const void* p) {
  return (unsigned)(size_t)p;
}

__global__ __launch_bounds__(NTHREADS) void steal_nms_main(
    const float* __restrict__ pred, const int* __restrict__ labels,
    float* __restrict__ partials) {
  __shared__ int      s_lab[HALO_N];        //  5440 B
  __shared__ u64      s_ind[HALO_N * 2];    // 21760 B  (lo,hi byte-packed one-hot)
  __shared__ float    s_exp[CG * HALO_N];   // 21760 B
  __shared__ unsigned s_dir[TH * TW];       //  4096 B  (2 bits/class * 16)
  __shared__ float    s_red[16];

  const int tid = threadIdx.x;
  const int b  = blockIdx.z;
  const int h0 = blockIdx.y * TH;
  const int w0 = blockIdx.x * TW;

  // ---- Phase 1: async-copy label halo tile into LDS (CDNA5 async-to-LDS) ----
  {
    const int* lab_base = labels + (size_t)b * H_IMG * W_IMG;
    for (int p = tid; p < HALO_N; p += NTHREADS) {
      int r = p / WS, cc = p - r * WS;
      int gh = clampi(h0 + r - 2, 0, H_IMG - 1);
      int gw = clampi(w0 + cc - 2, 0, W_IMG - 1);
      unsigned voff = (unsigned)((gh * W_IMG + gw) * 4);
      unsigned doff = lds_off(&s_lab[p]);
      asm volatile("global_load_async_to_lds_b32 %0, %1, %2"
                   :: "v"(doff), "v"(voff),
                      "s"((unsigned long long)(size_t)lab_base)
                   : "memory");
    }
    asm volatile("s_wait_asynccnt 0" ::: "memory");
  }
  __syncthreads();

  // ---- Phase 2: byte-packed one-hot indicator words (8 classes per u64) ----
  for (int p = tid; p < HALO_N; p += NTHREADS) {
    int lab = s_lab[p] & 15;
    s_ind[p * 2 + 0] = (lab < 8) ? (1ull << (8 * lab)) : 0ull;
    s_ind[p * 2 + 1] = (lab >= 8) ? (1ull << (8 * (lab - 8))) : 0ull;
  }
  __syncthreads();

  // ---- Phase 3: 16-class edge-direction classification per pixel ----
  // Composed separable 5x5 kernels (exact in the interior):
  //   gxx = (1/64)*[1,4,6,4,1]_v (x) [1,0,-2,0,1]_h
  //   gyy = (1/64)*[1,0,-2,0,1]_v (x) [1,4,6,4,1]_h
  //   gxy = (1/64)*[-1,-2,0,2,1] (x) [-1,-2,0,2,1]
  for (int pix = tid; pix < TH * TW; pix += NTHREADS) {
    int i = pix >> 6, j = pix & 63;
    u64 Pxx[2], Nxx[2], Pyy[2], Nyy[2], Pxy[2], Nxy[2];
#pragma unroll
    for (int wsel = 0; wsel < 2; ++wsel) {
      u64 pxx = 0, nxx = 0, pyy = 0, nyy = 0, pxy = 0, nxy = 0;
#pragma unroll
      for (int cc = 0; cc < 5; ++cc) {
        const u64* col = &s_ind[(i * WS + (j + cc)) * 2 + wsel];
        u64 i0 = col[0 * WS * 2], i1 = col[1 * WS * 2], i2 = col[2 * WS * 2],
            i3 = col[3 * WS * 2], i4 = col[4 * WS * 2];
        u64 cYp  = i0 + i4;                                    // V2 vertical, +part
        u64 cYn  = i2 << 1;                                    // V2 vertical, -part
        u64 cX   = cYp + ((i1 + i3) << 2) + (i2 << 2) + (i2 << 1); // U2 vertical
        u64 cXYp = (i3 << 1) + i4;                             // UV vertical, +part
        u64 cXYn = i0 + (i1 << 1);                             // UV vertical, -part
        // horizontal V2 = [1,0,-2,0,1]  -> gxx
        if (cc == 0 || cc == 4) pxx += cX;
        if (cc == 2) nxx += cX << 1;
        // horizontal U2 = [1,4,6,4,1]   -> gyy
        if (cc == 0 || cc == 4)      { pyy += cYp;                        nyy += cYn; }
        else if (cc == 2)            { pyy += (cYp << 2) + (cYp << 1);    nyy += (cYn << 2) + (cYn << 1); }
        else                         { pyy += cYp << 2;                   nyy += cYn << 2; }
        // horizontal UV = [-1,-2,0,2,1] -> gxy (sign-split)
        if (cc == 0) { pxy += cXYn;      nxy += cXYp; }
        if (cc == 1) { pxy += cXYn << 1; nxy += cXYp << 1; }
        if (cc == 3) { pxy += cXYp << 1; nxy += cXYn << 1; }
        if (cc == 4) { pxy += cXYp;      nxy += cXYn; }
      }
      Pxx[wsel] = pxx; Nxx[wsel] = nxx; Pyy[wsel] = pyy;
      Nyy[wsel] = nyy; Pxy[wsel] = pxy; Nxy[wsel] = nxy;
    }
    unsigned packed = 0;
#pragma unroll
    for (int c = 0; c < 16; ++c) {
      const int wsel = c >> 3, sh = (c & 7) * 8;
      int gxx = (int)((Pxx[wsel] >> sh) & 255) - (int)((Nxx[wsel] >> sh) & 255);
      int gyy = (int)((Pyy[wsel] >> sh) & 255) - (int)((Nyy[wsel] >> sh) & 255);
      int gxy = (int)((Pxy[wsel] >> sh) & 255) - (int)((Nxy[wsel] >> sh) & 255);
      // t = (gyy/64)*sign(eps-gxy/64) / (gxx/64 + eps) = s*gyy / (gxx + 64*eps)
      float num = (gxy <= 0) ? (float)gyy : -(float)gyy;
      float den = (float)gxx + 7.6293945e-6f;  // 64 * FLT_EPSILON
      // classification only compares t against constants: v_rcp_f32 is plenty
      float t = num * __builtin_amdgcn_rcpf(den);
      // round(atan(t)*5/pi): boundaries tan(pi/10), tan(3pi/10); k=-1 -> dir0, k=-2 -> dir3
      int dir = (t > 1.3763819f) ? 2 : (t > 0.32491970f) ? 1
              : (t < -1.3763819f) ? 3 : 0;
      packed |= (unsigned)dir << (2 * c);
    }
    s_dir[pix] = packed;
  }

  // ---- Phase 4: channel groups: exp(pred) tile in LDS + NMS accumulation ----
  float partial = 0.f;
  const float* pred_b = pred + (size_t)b * C_TOTAL * H_IMG * W_IMG;
  for (int g = 0; g < C_TOTAL / CG; ++g) {
    __syncthreads();  // protect s_exp reuse (and s_dir on first pass)
    for (int cl = 0; cl < CG; ++cl) {
      const float* pc = pred_b + (size_t)(g * CG + cl) * H_IMG * W_IMG;
      for (int p = tid; p < HALO_N; p += NTHREADS) {
        int r = p / WS, cc = p - r * WS;
        int gh = clampi(h0 + r - 2, 0, H_IMG - 1);
        int gw = clampi(w0 + cc - 2, 0, W_IMG - 1);
        s_exp[cl * HALO_N + p] = __expf(pc[gh * W_IMG + gw]);
      }
    }
    __syncthreads();
    for (int pix = tid; pix < TH * TW; pix += NTHREADS) {
      int i = pix >> 6, j = pix & 63;
      int gh = h0 + i, gw = w0 + j;
      if (gh < 2 || gh >= H_IMG - 2 || gw < 2 || gw >= W_IMG - 2) continue;
      unsigned dirw = s_dir[pix];
#pragma unroll
      for (int cl = 0; cl < CG; ++cl) {
        int c = g * CG + cl;
        int dir = (dirw >> (2 * c)) & 3;
        int p0 = (cl * HS + i + 2) * WS + (j + 2);
        // taps: dir0 (0,-2..1); dir2 (-2..1,0); dir1/3 (-2,1),(-1,0),(0,-1),(1,-2)
        bool diag = dir & 1;
        int t0 = (dir == 0) ? -2 : (diag ? (-2 * WS + 1) : -2 * WS);
        int t1 = (dir == 0) ? -1 : -WS;
        int t2 = diag ? -1 : 0;
        int t3 = (dir == 0) ? 1 : (diag ? (WS - 2) : WS);
        float e = s_exp[p0];
        float d = s_exp[p0 + t0] + s_exp[p0 + t1] + s_exp[p0 + t2] + s_exp[p0 + t3];
        // fast reciprocal (v_rcp_f32 + v_mul) instead of IEEE div sequence
        partial += e * __builtin_amdgcn_rcpf(d);
      }
    }
  }
  __syncthreads();

  // ---- Phase 5: exact wave reduction via V_WMMA_F32_16X16X4_F32, B = ones ----
  // A: lanes0-15 {k0=partial,k1=0}, lanes16-31 {k2=partial,k3=0}.
  // D[m,n] = partial[m] + partial[m+16] (same for all n). Sum of a lane's 8 D
  // regs = rows 0..7 (lanes<16) or rows 8..15 (lanes>=16) -> lane0+lane16 = wave sum.
  v2f a;  a.x = partial; a.y = 0.f;
  v2f bo; bo.x = 1.f;    bo.y = 1.f;
  v8f acc = {};
  acc = __builtin_amdgcn_wmma_f32_16x16x4_f32(false, a, false, bo, (short)0, acc,
                                              false, false);
  float half_sum = acc[0] + acc[1] + acc[2] + acc[3] + acc[4] + acc[5] + acc[6] + acc[7];
  int lane = tid & 31, wave = tid >> 5;
  if (lane == 0)  s_red[wave * 2 + 0] = half_sum;
  if (lane == 16) s_red[wave * 2 + 1] = half_sum;
  __syncthreads();
  if (tid == 0) {
    float s = 0.f;
#pragma unroll
    for (int k = 0; k < 16; ++k) s += s_red[k];
    partials[((size_t)blockIdx.z * gridDim.y + blockIdx.y) * gridDim.x + blockIdx.x] = s;
  }
}

// Deterministic fixed-order final reduction of block partials.
__global__ __launch_bounds__(NTHREADS) void steal_nms_reduce(
    const float* __restrict__ partials, float* __restrict__ out, int n) {
  __shared__ float r[NTHREADS];
  float a = 0.f;
  for (int i = threadIdx.x; i < n; i += NTHREADS) a += partials[i];
  r[threadIdx.x] = a;
  __syncthreads();
  for (int s = NTHREADS / 2; s > 0; s >>= 1) {
    if (threadIdx.x < s) r[threadIdx.x] += r[threadIdx.x + s];
    __syncthreads();
  }
  if (threadIdx.x == 0) out[0] = r[0];
}

extern "C" void kernel_launch(void* const* d_in, const int* in_sizes, int n_in,
                              void* d_out, int out_size, void* d_ws, size_t ws_size,
                              hipStream_t stream) {
  const float* pred = (const float*)d_in[0];
  const int* labels = (const int*)d_in[1];   // harness delivers integer input as int32
  int B = in_sizes[1] / (H_IMG * W_IMG);     // 4
  dim3 grid(W_IMG / TW, H_IMG / TH, B);      // 16 x 32 x 4 = 2048 blocks
  float* part = (float*)d_ws;
  steal_nms_main<<<grid, NTHREADS, 0, stream>>>(pred, labels, part);
  int n = B * (H_IMG / TH) * (W_IMG / TW);
  steal_nms_reduce<<<1, NTHREADS, 0, stream>>>(part, (float*)d_out, n);
}